// ListMLE_56367150792862
// MI455X (gfx1250) — compile-verified
//
#include <hip/hip_runtime.h>
#include <stdint.h>

// ListMLE forward for MI455X (gfx1250).
// Per (b,h) row: stable sort by input2 (masked -> +1e5), gather input1 (4
// choices), exp, masked cumsum, product of (p+eps)/(cum+eps).
// Output (bs, nh, 4) f32.
//
// CDNA5 paths used: global_load_async_to_lds_b128 + s_wait_asynccnt to stage
// the sort keys / mask rows straight into LDS (the bitonic sort needs them in
// LDS anyway). WMMA is not applicable: no matmul in this workload; it is
// bandwidth-bound (~192 MB @ 23.3 TB/s ≈ 8 us floor, input1 = 128 MB of it).
//
// This revision: (a) all 4 choices' gathers issued as one 32-load burst per
// thread to maximize MLP on the dominant input1 stream; (b) the workgroup
// scan/reduce runs once on a float4 payload (all 4 choices at once), cutting
// barriers from ~170 to ~25 beyond the sort's 66.

#define NELEM 2048
#define NTHREADS 256
#define EPT (NELEM / NTHREADS)   // 8 elements per thread
#define EPSV 1e-9f
#define MASK_FILL 100000.0f

// ---- CDNA5 async global->LDS copy (per-lane 16B), inline asm ----
__device__ __forceinline__ void async_g2l_b128(uint32_t lds_byte_addr, uint64_t gaddr) {
  asm volatile("global_load_async_to_lds_b128 %0, %1, off"
               :: "v"(lds_byte_addr), "v"(gaddr)
               : "memory");
}
__device__ __forceinline__ void wait_async0() {
  asm volatile("s_wait_asynccnt 0" ::: "memory");
}

__global__ __launch_bounds__(NTHREADS) void listmle_fwd_kernel(
    const float* __restrict__ input1,   // (bs, 4, nh, n)
    const float* __restrict__ input2,   // (bs, 1, nh, n)
    const int*   __restrict__ mask2,    // (bs, 1, nh, n)
    float*       __restrict__ out,      // (bs, nh, 4)
    int nh) {
  __shared__ __align__(16) unsigned long long kv[NELEM]; // 16 KB sort keys
  __shared__ __align__(16) float  sval[NELEM];           // 8 KB input2 row
  __shared__ __align__(16) int    smask[NELEM];          // 8 KB mask2 row
  __shared__ __align__(16) float4 red4[NTHREADS];        // 4 KB scan/reduce (4 choices)

  const int tid = threadIdx.x;
  const int row = blockIdx.x;            // b*nh + h
  const int b   = row / nh;
  const int h   = row - b * nh;

  // ---- Stage input2 + mask2 rows into LDS via the async copy engine ----
  {
    const size_t rowoff = (size_t)row * NELEM;
    uint64_t gv = (uint64_t)(size_t)(input2 + rowoff);
    uint64_t gm = (uint64_t)(size_t)(mask2  + rowoff);
    uint32_t lv = (uint32_t)(size_t)&sval[0];   // low 32 bits of generic ptr == LDS offset
    uint32_t lm = (uint32_t)(size_t)&smask[0];
    uint32_t o0 = (uint32_t)tid * 16u;          // 256 lanes * 16B = 4KB per issue
    uint32_t o1 = o0 + 4096u;
    async_g2l_b128(lv + o0, gv + o0);
    async_g2l_b128(lv + o1, gv + o1);
    async_g2l_b128(lm + o0, gm + o0);
    async_g2l_b128(lm + o1, gm + o1);
    wait_async0();                              // own wave's portion complete
  }
  __syncthreads();                              // cross-wave visibility

  // ---- Build stable-sort keys: (sortable_f32 << 32) | index ----
  for (int i = tid; i < NELEM; i += NTHREADS) {
    float f = (smask[i] > 0) ? sval[i] : MASK_FILL;
    uint32_t u = __float_as_uint(f);
    uint32_t s = (u & 0x80000000u) ? ~u : (u | 0x80000000u); // ascending-float order
    kv[i] = ((unsigned long long)s << 32) | (uint32_t)i;     // index = stable tiebreak
  }
  __syncthreads();

  // ---- Bitonic sort (ascending) of 2048 u64 keys in LDS ----
  for (uint32_t k = 2; k <= (uint32_t)NELEM; k <<= 1) {
    for (uint32_t j = k >> 1; j > 0; j >>= 1) {
      #pragma unroll
      for (int it = 0; it < NELEM / 2 / NTHREADS; ++it) {
        uint32_t p = (uint32_t)tid + (uint32_t)it * NTHREADS;    // p in [0, N/2)
        uint32_t i = ((p & ~(j - 1u)) << 1) | (p & (j - 1u));    // bit j clear
        uint32_t q = i | j;
        bool up = ((i & k) == 0);
        unsigned long long a = kv[i];
        unsigned long long c = kv[q];
        if ((a > c) == up) { kv[i] = c; kv[q] = a; }
      }
      __syncthreads();
    }
  }

  // ---- Sorted indices + masks for this thread's contiguous chunk ----
  const uint32_t base = (uint32_t)tid * EPT;
  uint32_t idx8[EPT];
  float    m8[EPT];
  #pragma unroll
  for (int e = 0; e < EPT; ++e) {
    idx8[e] = (uint32_t)(kv[base + e] & 0xffffffffu);
    m8[e]   = (smask[idx8[e]] > 0) ? 1.0f : 0.0f;
  }

  // ---- One 32-load gather burst: all 4 choices of input1 ----
  const size_t rowbase = ((size_t)(b * 4) * nh + h) * (size_t)NELEM;
  const size_t cstride = (size_t)nh * (size_t)NELEM;   // choice stride in elements
  float xg[4][EPT];
  #pragma unroll
  for (int c = 0; c < 4; ++c) {
    const float* g1 = input1 + rowbase + (size_t)c * cstride;
    #pragma unroll
    for (int e = 0; e < EPT; ++e) xg[c][e] = g1[idx8[e]];
  }

  // ---- exp + per-thread running sums (all choices) ----
  float p4[4][EPT];
  float run[4];
  #pragma unroll
  for (int c = 0; c < 4; ++c) {
    float r = 0.0f;
    #pragma unroll
    for (int e = 0; e < EPT; ++e) {
      float pe = expf(xg[c][e]) * m8[e];     // ALPHA == 1.0
      r += pe;
      p4[c][e] = pe;
    }
    run[c] = r;
  }

  // ---- Workgroup inclusive scan of thread totals, 4 choices as float4 ----
  red4[tid] = make_float4(run[0], run[1], run[2], run[3]);
  __syncthreads();
  float4 inc = make_float4(run[0], run[1], run[2], run[3]);
  for (int off = 1; off < NTHREADS; off <<= 1) {
    float4 t = make_float4(0.f, 0.f, 0.f, 0.f);
    if (tid >= off) t = red4[tid - off];
    __syncthreads();
    inc.x += t.x; inc.y += t.y; inc.z += t.z; inc.w += t.w;
    red4[tid] = inc;
    __syncthreads();
  }
  float excl[4] = { inc.x - run[0], inc.y - run[1], inc.z - run[2], inc.w - run[3] };

  // ---- Per-element terms + per-thread product (recompute running sums) ----
  float prod[4];
  #pragma unroll
  for (int c = 0; c < 4; ++c) {
    float r = 0.0f;
    float pr = 1.0f;
    #pragma unroll
    for (int e = 0; e < EPT; ++e) {
      r += p4[c][e];
      float cs = (excl[c] + r) * m8[e];      // proj_cum = cumsum * mask
      pr *= (p4[c][e] + EPSV) / (cs + EPSV); // masked entries -> eps/eps = 1
    }
    prod[c] = pr;
  }

  // ---- Workgroup product reduction (float4), single b128 store ----
  red4[tid] = make_float4(prod[0], prod[1], prod[2], prod[3]);
  __syncthreads();
  for (int off = NTHREADS >> 1; off > 0; off >>= 1) {
    if (tid < off) {
      float4 a = red4[tid], c2 = red4[tid + off];
      a.x *= c2.x; a.y *= c2.y; a.z *= c2.z; a.w *= c2.w;
      red4[tid] = a;
    }
    __syncthreads();
  }
  if (tid == 0) {
    // out[(b*nh + h)*4 + c], c contiguous -> 16B aligned vector store
    *(float4*)(out + (size_t)row * 4) = red4[0];
  }
}

extern "C" void kernel_launch(void* const* d_in, const int* in_sizes, int n_in,
                              void* d_out, int out_size, void* d_ws, size_t ws_size,
                              hipStream_t stream) {
  // setup_inputs order: input1 (f32), mask1 (i32, unused), input2 (f32), mask2 (i32)
  const float* input1 = (const float*)d_in[0];
  const float* input2 = (const float*)d_in[2];
  const int*   mask2  = (const int*)d_in[3];
  float* out = (float*)d_out;

  const int nh = 32;
  const int n  = NELEM;
  const int bs = in_sizes[2] / (nh * n);     // in_sizes[2] = bs*1*nh*n

  dim3 grid(bs * nh);
  dim3 block(NTHREADS);
  listmle_fwd_kernel<<<grid, block, 0, stream>>>(input1, input2, mask2, out, nh);
}